// DiscreteDosePKPDModel_59416577573265
// MI455X (gfx1250) — compile-verified
//
#include <hip/hip_runtime.h>
#include <math.h>

// ---------------------------------------------------------------------------
// PK/PD RK4 scan for MI455X (gfx1250, wave32).
//  Kernel 1: per-subject parameter head  params = softplus(feats@W + b)+0.01
//            done as V_WMMA_F32_16X16X4_F32, one WMMA per 16 subjects.
//  Kernel 2: sequential RK4 scan, 1 thread = 1 subject, LDS-staged coalesced
//            trajectory writes (16 steps per flush, 256B bursts per subject).
// ---------------------------------------------------------------------------

typedef __attribute__((ext_vector_type(2))) float v2f;
typedef __attribute__((ext_vector_type(8))) float v8f;

#define NSTEPS 2048
#define NROWS  2049   // NSTEPS + 1
#define CHUNK  16

// ---------------------------- Kernel 1: WMMA head ---------------------------
// A (16x4 f32, M=subjects, K=features padded to 4):
//   lanes 0-15 : v0 = A[M=lane][K=0], v1 = A[lane][K=1]
//   lanes 16-31: v0 = A[lane-16][K=2], v1 = A[lane-16][K=3] (=0)
// B (4x16 f32, K x N=params padded to 16): symmetric striping
//   lanes 0-15 : v0 = B[K=0][N=lane], v1 = B[K=1][N=lane]
//   lanes 16-31: v0 = B[K=2][N=lane-16], v1 = B[K=3][..] (=0)
// C/D (16x16 f32, 8 VGPRs): vgpr r -> M = r + 8*(lane>=16), N = lane&15
__global__ __launch_bounds__(32)
void pkpd_params_wmma(const float* __restrict__ cov,        // [N][2] bw, comed
                      const float* __restrict__ dose_int,   // [N]
                      const float* __restrict__ W,          // [3][9]
                      const float* __restrict__ bias,       // [9]
                      float* __restrict__ params,           // [9][N] (ws)
                      int N)
{
    const int lane = threadIdx.x;        // 0..31
    const int half = lane >> 4;          // 0 | 1
    const int l    = lane & 15;
    const int tile = blockIdx.x;         // N/16 tiles
    const int subj = tile * 16 + l;

    // A fragment: feats = [bw*0.01, comed, dose_intensity, 0]
    v2f a;
    if (half == 0) {
        a.x = cov[subj * 2 + 0] * 0.01f;
        a.y = cov[subj * 2 + 1];
    } else {
        a.x = dose_int[subj];
        a.y = 0.0f;
    }

    // B fragment + bias (columns >= 9 are zero padding)
    v2f bf; float bv;
    if (l < 9) {
        if (half == 0) { bf.x = W[0 * 9 + l]; bf.y = W[1 * 9 + l]; }
        else           { bf.x = W[2 * 9 + l]; bf.y = 0.0f;        }
        bv = bias[l];
    } else {
        bf.x = 0.0f; bf.y = 0.0f; bv = 0.0f;
    }

    v8f c;
#pragma unroll
    for (int r = 0; r < 8; ++r) c[r] = bv;   // C[m][n] = b[n]

    // D = A*B + C   -> v_wmma_f32_16x16x4_f32
    v8f d = __builtin_amdgcn_wmma_f32_16x16x4_f32(
        /*neg_a=*/false, a, /*neg_b=*/false, bf,
        /*c_mod=*/(short)0, c, /*reuse_a=*/false, /*reuse_b=*/false);

    if (l < 9) {
#pragma unroll
        for (int r = 0; r < 8; ++r) {
            const int m = half * 8 + r;                 // subject row in tile
            const float x  = d[r];
            const float sp = fmaxf(x, 0.0f) + log1pf(expf(-fabsf(x)));
            params[l * N + tile * 16 + m] = sp + 0.01f;
        }
    }
}

// ---------------------------- Kernel 2: RK4 scan ----------------------------
struct S4 { float ad, ac, ap, r; };

__device__ __forceinline__ S4 rhs(const S4 y, float Ka, float kc_sum,
                                  float k12, float k21, float invVc,
                                  float Kin, float Kout, float Imax, float IC50)
{
    S4 d;
    d.ad = -Ka * y.ad;
    d.ac = Ka * y.ad - kc_sum * y.ac + k21 * y.ap;
    d.ap = k12 * y.ac - k21 * y.ap;
    const float conc  = y.ac * invVc;
    const float inhib = (Imax * conc) / (IC50 + conc + 1e-6f);
    d.r  = Kin * (1.0f - inhib) - Kout * y.r;
    return d;
}

__device__ __forceinline__ S4 axpy(const S4 y, float h, const S4 k)
{
    S4 o;
    o.ad = fmaf(h, k.ad, y.ad);
    o.ac = fmaf(h, k.ac, y.ac);
    o.ap = fmaf(h, k.ap, y.ap);
    o.r  = fmaf(h, k.r,  y.r);
    return o;
}

__global__ __launch_bounds__(32)
void pkpd_scan(const float* __restrict__ params,       // [9][N]
               const float* __restrict__ dose_amounts, // [N][8]
               float* __restrict__ out,                // [N][2049][4]
               int N)
{
    const int lane  = threadIdx.x;          // 0..31
    const int subj0 = blockIdx.x * 32;
    const int subj  = subj0 + lane;

    // 9 parameters (coalesced loads: [9][N])
    const float Ka   = params[0 * N + subj];
    const float CL   = params[1 * N + subj];
    const float Vc   = params[2 * N + subj];
    const float Q    = params[3 * N + subj];
    const float Vp   = params[4 * N + subj];
    const float Kin  = params[5 * N + subj];
    const float Kout = params[6 * N + subj];
    const float Imax = params[7 * N + subj];
    const float IC50 = params[8 * N + subj];

    // loop-invariant rate constants (1 divide per RHS eval remains)
    const float invVc  = 1.0f / Vc;
    const float ke     = CL * invVc;
    const float k12    = Q * invVc;
    const float k21    = Q / Vp;
    const float kc_sum = ke + k12;

    const float dt  = 504.0f / 2048.0f;     // 0.24609375 (exact)
    const float hdt = 0.5f * dt;
    const float dt6 = dt * (1.0f / 6.0f);

    S4 y; y.ad = 0.0f; y.ac = 0.0f; y.ap = 0.0f; y.r = 16.0f;

    float4* __restrict__ out4 = (float4*)out;
    // row 0 = y0 (pre-dose)
    out4[(size_t)subj * NROWS] = make_float4(y.ad, y.ac, y.ap, y.r);

    // 17-slot pad on the step axis breaks the 32-way bank conflict
    __shared__ float4 stage[32][CHUNK + 1];

    for (int s = 0; s < NSTEPS; ++s) {
        if ((s & 255) == 0)
            y.ad += dose_amounts[subj * 8 + (s >> 8)];

        const S4 k1 = rhs(y, Ka, kc_sum, k12, k21, invVc, Kin, Kout, Imax, IC50);
        const S4 y2 = axpy(y, hdt, k1);
        const S4 k2 = rhs(y2, Ka, kc_sum, k12, k21, invVc, Kin, Kout, Imax, IC50);
        const S4 y3 = axpy(y, hdt, k2);
        const S4 k3 = rhs(y3, Ka, kc_sum, k12, k21, invVc, Kin, Kout, Imax, IC50);
        const S4 y4 = axpy(y, dt, k3);
        const S4 k4 = rhs(y4, Ka, kc_sum, k12, k21, invVc, Kin, Kout, Imax, IC50);

        y.ad += dt6 * (k1.ad + 2.0f * (k2.ad + k3.ad) + k4.ad);
        y.ac += dt6 * (k1.ac + 2.0f * (k2.ac + k3.ac) + k4.ac);
        y.ap += dt6 * (k1.ap + 2.0f * (k2.ap + k3.ap) + k4.ap);
        y.r  += dt6 * (k1.r  + 2.0f * (k2.r  + k3.r ) + k4.r );

        stage[lane][s & (CHUNK - 1)] = make_float4(y.ad, y.ac, y.ap, y.r);

        if ((s & (CHUNK - 1)) == (CHUNK - 1)) {
            __syncthreads();                         // single-wave: fence only
            const int base_row = s - (CHUNK - 2);    // = s-14 -> rows s-14..s+1
#pragma unroll
            for (int k = 0; k < CHUNK; ++k) {
                const int i  = lane + k * 32;
                const int si = i >> 4;               // subject within wave
                const int st = i & (CHUNK - 1);      // step within chunk
                const float4 v = stage[si][st];
                out4[(size_t)(subj0 + si) * NROWS + base_row + st] = v;
            }
            __syncthreads();
        }
    }
}

// ------------------------------- launcher -----------------------------------
extern "C" void kernel_launch(void* const* d_in, const int* in_sizes, int n_in,
                              void* d_out, int out_size, void* d_ws, size_t ws_size,
                              hipStream_t stream)
{
    const float* cov      = (const float*)d_in[0]; // [N][2]
    const float* dose_int = (const float*)d_in[1]; // [N]
    const float* W        = (const float*)d_in[2]; // [3][9]
    const float* bias     = (const float*)d_in[3]; // [9]
    const float* doses    = (const float*)d_in[4]; // [N][8]
    float*       out      = (float*)d_out;         // [N][2049][4]
    float*       params   = (float*)d_ws;          // [9][N]

    const int N = in_sizes[1];                     // 4096

    pkpd_params_wmma<<<N / 16, 32, 0, stream>>>(cov, dose_int, W, bias, params, N);
    pkpd_scan<<<N / 32, 32, 0, stream>>>(params, doses, out, N);
}